// SentimentRNN_50740743635340
// MI455X (gfx1250) — compile-verified
//
#include <hip/hip_runtime.h>
#include <hip/hip_bf16.h>
#include <math.h>

#define VOCAB  400000
#define EMB    300
#define HIDDEN 128
#define OUT    3
#define SEQ    4096

#define EMB_PAD 304   // LDS row stride for the 16xEMB embedding tile

typedef __attribute__((ext_vector_type(2))) float v2f;
typedef __attribute__((ext_vector_type(8))) float v8f;

// ---------------------------------------------------------------------------
// Kernel 1: fused embedding gather + xproj GEMM via v_wmma_f32_16x16x4_f32.
// Grid: SEQ/16 = 256 blocks of 256 threads (8 waves).
// Each block: gather 16 embedding rows into LDS, then wave w computes the
// 16x16 output tile at columns [16w, 16w+16) with a 75-step K=4 WMMA chain.
// ---------------------------------------------------------------------------
__global__ __launch_bounds__(256) void xproj_wmma_kernel(
    const int*   __restrict__ X,
    const float* __restrict__ Wemb,   // [VOCAB, EMB]
    const float* __restrict__ W_ih,   // [HIDDEN, EMB]
    const float* __restrict__ b_ih,   // [HIDDEN]
    float*       __restrict__ xproj)  // [SEQ, HIDDEN]
{
    __shared__ float at[16 * EMB_PAD];

    const int tid     = threadIdx.x;
    const int rowBase = blockIdx.x * 16;

    // --- gather 16 embedding rows: 16 threads per row, coalesced along EMB ---
    {
        const int r  = tid >> 4;          // 0..15 : which row of the tile
        const int c0 = tid & 15;
        const long long src = (long long)X[rowBase + r] * EMB;
        for (int c = c0; c < EMB; c += 16)
            at[r * EMB_PAD + c] = Wemb[src + c];
    }
    __syncthreads();

    // --- WMMA: D[16x16] = A[16x300] * B[300x16] over 75 K=4 steps ---
    const int wave    = tid >> 5;         // 0..7 -> which 16-col tile
    const int lane    = tid & 31;
    const int colBase = wave * 16;
    const int mn      = lane & 15;        // M for A, N for B/D
    const int khalf   = (lane >> 4) * 2;  // lanes 16..31 hold K+2,K+3

    const float* arow = at   + mn * EMB_PAD + khalf;                       // A[mn][k+khalf]
    const float* brow = W_ih + (long long)(colBase + mn) * EMB + khalf;    // B[k][n] = W_ih[n][k]

    v8f c = {};
    #pragma unroll 5
    for (int k0 = 0; k0 < EMB; k0 += 4) {
        v2f a, b;
        a.x = arow[k0];  a.y = arow[k0 + 1];
        b.x = brow[k0];  b.y = brow[k0 + 1];
        // (neg_a, A, neg_b, B, c_mod, C, reuse_a, reuse_b)
        c = __builtin_amdgcn_wmma_f32_16x16x4_f32(false, a, false, b,
                                                  (short)0, c, false, false);
    }

    // --- store D + bias: VGPR j -> M=j (lanes 0-15) / M=8+j (lanes 16-31) ---
    const int   n     = colBase + mn;
    const float bias  = b_ih[n];
    const int   mBase = rowBase + ((lane >> 4) << 3);
    #pragma unroll
    for (int j = 0; j < 8; ++j)
        xproj[(mBase + j) * HIDDEN + n] = c[j] + bias;
}

// ---------------------------------------------------------------------------
// Kernel 2: serial Elman recurrence + final FC + log-softmax.
// One block of 128 threads; thread t owns h[t]. Its row of W_hh lives in
// 128 VGPRs; h lives in LDS, DOUBLE-BUFFERED so each of the 4096 serial
// steps needs only ONE barrier:
//   h[next][t] = tanh(xproj[step][t] + b_hh[t] + sum_k W_hh[t][k] * h[cur][k])
// ---------------------------------------------------------------------------
__global__ __launch_bounds__(128) void rnn_recurrence_kernel(
    const float* __restrict__ xproj,  // [SEQ, HIDDEN]
    const float* __restrict__ W_hh,   // [HIDDEN, HIDDEN]
    const float* __restrict__ b_hh,   // [HIDDEN]
    const float* __restrict__ W_fc,   // [OUT, HIDDEN]
    const float* __restrict__ b_fc,   // [OUT]
    float*       __restrict__ out)    // [OUT]
{
    __shared__ float h[2][HIDDEN];
    __shared__ float logits[OUT];

    const int tid = threadIdx.x;

    // Register-resident row of W_hh (out[t] = sum_k W_hh[t][k] * h[k]).
    float w[HIDDEN];
    #pragma unroll
    for (int k = 0; k < HIDDEN; ++k) w[k] = W_hh[tid * HIDDEN + k];
    const float bias = b_hh[tid];

    h[0][tid] = 0.0f;   // h0 = zeros
    __syncthreads();

    for (int t = 0; t < SEQ; ++t) {
        const int cur = t & 1;

        // warm L0/L2 for an upcoming step (global_prefetch_b8)
        const int tp = (t + 8 < SEQ) ? (t + 8) : (SEQ - 1);
        __builtin_prefetch(xproj + tp * HIDDEN + tid, 0, 1);

        float acc = xproj[t * HIDDEN + tid] + bias;
        #pragma unroll
        for (int k = 0; k < HIDDEN; ++k)
            acc = fmaf(w[k], h[cur][k], acc);

        // publish into the other buffer; current buffer stays intact for
        // any thread still reading it, so a single barrier suffices.
        h[cur ^ 1][tid] = tanhf(acc);
        __syncthreads();
    }

    // Final FC on h[SEQ&1] (the buffer written at t = SEQ-1).
    const float* hT = h[SEQ & 1];
    if (tid < OUT) {
        float acc = b_fc[tid];
        for (int k = 0; k < HIDDEN; ++k)
            acc = fmaf(W_fc[tid * HIDDEN + k], hT[k], acc);
        logits[tid] = acc;
    }
    __syncthreads();

    if (tid == 0) {
        const float m  = fmaxf(logits[0], fmaxf(logits[1], logits[2]));
        const float s  = expf(logits[0] - m) + expf(logits[1] - m) + expf(logits[2] - m);
        const float ls = logf(s);
        out[0] = logits[0] - m - ls;
        out[1] = logits[1] - m - ls;
        out[2] = logits[2] - m - ls;
    }
}

// ---------------------------------------------------------------------------
extern "C" void kernel_launch(void* const* d_in, const int* in_sizes, int n_in,
                              void* d_out, int out_size, void* d_ws, size_t ws_size,
                              hipStream_t stream) {
    (void)in_sizes; (void)n_in; (void)out_size; (void)ws_size;

    const int*   X     = (const int*)  d_in[0];
    const float* Wemb  = (const float*)d_in[1];
    const float* W_ih  = (const float*)d_in[2];
    const float* b_ih  = (const float*)d_in[3];
    const float* W_hh  = (const float*)d_in[4];
    const float* b_hh  = (const float*)d_in[5];
    const float* W_fc  = (const float*)d_in[6];
    const float* b_fc  = (const float*)d_in[7];
    float*       outp  = (float*)d_out;
    float*       xproj = (float*)d_ws;            // SEQ*HIDDEN*4 = 2 MB scratch

    xproj_wmma_kernel<<<SEQ / 16, 256, 0, stream>>>(X, Wemb, W_ih, b_ih, xproj);
    rnn_recurrence_kernel<<<1, 128, 0, stream>>>(xproj, W_hh, b_hh, W_fc, b_fc, outp);
}